// _BaseRelGraphConv_86938728005789
// MI455X (gfx1250) — compile-verified
//
#include <hip/hip_runtime.h>

typedef float v2f __attribute__((ext_vector_type(2)));
typedef float v8f __attribute__((ext_vector_type(8)));

#define D 64
#define NRELS 8

// ---------------------------------------------------------------------------
// Kernel 1: WMMA fp32 GEMM.
//   grid.x = node tiles (16 rows each), grid.y = matrix index k in [0..8]
//   k < 8  : hw[n][k][:] = h[n] @ W[k]            (written to workspace)
//   k == 8 : out[n][:]   = h[n] @ loopW + h_bias  (output accumulator init)
// block = 128 threads (4 waves); wave w computes output columns [16w, 16w+16).
// Each 16x16 output tile accumulates 16 x V_WMMA_F32_16X16X4_F32 over K=64.
// ---------------------------------------------------------------------------
__global__ void rgcn_gemm_wmma(const float* __restrict__ h,
                               const float* __restrict__ W,      // (8,64,64)
                               const float* __restrict__ loopW,  // (64,64)
                               const float* __restrict__ h_bias, // (64)
                               float* __restrict__ hw,           // (N,8,64) or null
                               float* __restrict__ out,          // (N,64)
                               int n_nodes, int only_loop) {
    const int tile = blockIdx.x;
    const int k    = only_loop ? NRELS : (int)blockIdx.y;
    const int wave = threadIdx.x >> 5;   // 0..3 -> column tile
    const int lane = threadIdx.x & 31;
    const int lo   = lane & 15;          // row / col within half
    const int hi   = lane >> 4;          // 0 or 1 (selects K pair / row half)

    // A fragment rows: lane holds row (tile*16 + lo); clamp for safety.
    int row_a = tile * 16 + lo;
    if (row_a >= n_nodes) row_a = n_nodes - 1;
    const float* __restrict__ A = h + (size_t)row_a * D;

    const float* __restrict__ B = (k < NRELS) ? (W + (size_t)k * D * D) : loopW;
    const int colb = wave * 16;
    const int col  = colb + lo;

    v8f acc = {};
#pragma unroll
    for (int s = 0; s < 16; ++s) {
        const int k0 = 4 * s;
        // A 16x4 f32 fragment: lanes 0-15 -> K = k0,k0+1 ; lanes 16-31 -> K = k0+2,k0+3
        v2f a;
        a.x = A[k0 + hi * 2];
        a.y = A[k0 + hi * 2 + 1];
        // B 4x16 f32 fragment: VGPR0 rows k0 (lanes 0-15) / k0+2 (lanes 16-31),
        //                      VGPR1 rows k0+1 / k0+3 ; lanes index columns.
        v2f b;
        b.x = B[(size_t)(k0 + hi * 2)     * D + col];
        b.y = B[(size_t)(k0 + hi * 2 + 1) * D + col];
        acc = __builtin_amdgcn_wmma_f32_16x16x4_f32(
            /*neg_a=*/false, a, /*neg_b=*/false, b,
            /*c_mod=*/(short)0, acc, /*reuse_a=*/false, /*reuse_b=*/false);
    }

    // C/D layout: VGPR v, lanes 0-15 -> row M=v ; lanes 16-31 -> row M=v+8.
    if (k < NRELS) {
#pragma unroll
        for (int v = 0; v < 8; ++v) {
            const int m = tile * 16 + hi * 8 + v;
            if (m < n_nodes)
                hw[((size_t)m * NRELS + k) * D + col] = acc[v];
        }
    } else {
        const float bias = h_bias[col];
#pragma unroll
        for (int v = 0; v < 8; ++v) {
            const int m = tile * 16 + hi * 8 + v;
            if (m < n_nodes)
                out[(size_t)m * D + col] = acc[v] + bias;
        }
    }
}

// ---------------------------------------------------------------------------
// Kernel 2a: edge scatter from precomputed hw (hw is L2-resident: 102.4 MB
// fits in the 192 MB L2). One wave per edge; each lane handles 2 columns.
// ---------------------------------------------------------------------------
__global__ void rgcn_edge_hw(const float* __restrict__ hw,
                             const float* __restrict__ norm,
                             const int* __restrict__ src,
                             const int* __restrict__ dst,
                             const int* __restrict__ rel,
                             float* __restrict__ out, int n_edges) {
    const int t = blockIdx.x * blockDim.x + threadIdx.x;
    const int e = t >> 5;
    if (e >= n_edges) return;
    const int lane = threadIdx.x & 31;

    const int s = src[e];
    const int d = dst[e];
    const int r = rel[e];
    const float nm = norm[e];

    const float2 v = *(const float2*)(hw + ((size_t)s * NRELS + r) * D + lane * 2);
    float* o = out + (size_t)d * D + lane * 2;
    atomicAdd(o,     v.x * nm);
    atomicAdd(o + 1, v.y * nm);
}

// ---------------------------------------------------------------------------
// Kernel 2b (fallback if workspace too small): direct per-edge matvec.
// 64 threads per edge; W (128 KB) stays hot in cache.
// ---------------------------------------------------------------------------
__global__ void rgcn_edge_direct(const float* __restrict__ h,
                                 const float* __restrict__ W,
                                 const float* __restrict__ norm,
                                 const int* __restrict__ src,
                                 const int* __restrict__ dst,
                                 const int* __restrict__ rel,
                                 float* __restrict__ out, int n_edges) {
    const int t = blockIdx.x * blockDim.x + threadIdx.x;
    const int e = t >> 6;
    if (e >= n_edges) return;
    const int o = t & 63;

    const int s = src[e];
    const int d = dst[e];
    const int r = rel[e];
    const float nm = norm[e];

    const float* __restrict__ hv = h + (size_t)s * D;
    const float* __restrict__ Wp = W + (size_t)r * D * D + o;
    float acc = 0.f;
#pragma unroll
    for (int kk = 0; kk < D; ++kk)
        acc += hv[kk] * Wp[(size_t)kk * D];
    atomicAdd(&out[(size_t)d * D + o], acc * nm);
}

// ---------------------------------------------------------------------------
// Kernel 3: in-place ReLU on the accumulator.
// ---------------------------------------------------------------------------
__global__ void rgcn_relu(float* __restrict__ out, int n) {
    const int i = blockIdx.x * blockDim.x + threadIdx.x;
    if (i < n) out[i] = fmaxf(out[i], 0.0f);
}

extern "C" void kernel_launch(void* const* d_in, const int* in_sizes, int n_in,
                              void* d_out, int out_size, void* d_ws, size_t ws_size,
                              hipStream_t stream) {
    const float* h      = (const float*)d_in[0];
    const float* norm   = (const float*)d_in[1];
    const float* W      = (const float*)d_in[2];
    const float* loopW  = (const float*)d_in[3];
    const float* h_bias = (const float*)d_in[4];
    const int*   src    = (const int*)d_in[5];
    const int*   dst    = (const int*)d_in[6];
    const int*   rel    = (const int*)d_in[7];
    float* out = (float*)d_out;

    const int n_nodes = in_sizes[0] / D;
    const int n_edges = in_sizes[5];
    const int tiles   = (n_nodes + 15) / 16;

    const size_t hw_bytes = (size_t)n_nodes * NRELS * D * sizeof(float);
    const bool use_hw = (ws_size >= hw_bytes);
    float* hw = (float*)d_ws;

    if (use_hw) {
        // K1: hw for 8 relations + self-loop init of out (grid.y = 9)
        dim3 grid(tiles, NRELS + 1, 1);
        rgcn_gemm_wmma<<<grid, 128, 0, stream>>>(h, W, loopW, h_bias, hw, out,
                                                 n_nodes, /*only_loop=*/0);
        // K2: one wave per edge
        const long long threads = (long long)n_edges * 32;
        rgcn_edge_hw<<<(unsigned)((threads + 255) / 256), 256, 0, stream>>>(
            hw, norm, src, dst, rel, out, n_edges);
    } else {
        // K1: self-loop init only
        dim3 grid(tiles, 1, 1);
        rgcn_gemm_wmma<<<grid, 128, 0, stream>>>(h, W, loopW, h_bias, nullptr, out,
                                                 n_nodes, /*only_loop=*/1);
        // K2: direct per-edge matvec (64 threads per edge)
        const long long threads = (long long)n_edges * 64;
        rgcn_edge_direct<<<(unsigned)((threads + 255) / 256), 256, 0, stream>>>(
            h, W, norm, src, dst, rel, out, n_edges);
    }

    // K3: ReLU
    const int n = n_nodes * D;
    rgcn_relu<<<(n + 255) / 256, 256, 0, stream>>>(out, n);
}